// GCN_12979391169305
// MI455X (gfx1250) — compile-verified
//
#include <hip/hip_runtime.h>
#include <hip/hip_bf16.h>

typedef __attribute__((ext_vector_type(2))) float v2f;
typedef __attribute__((ext_vector_type(8))) float v8f;

static __device__ __forceinline__ float atomicAddF(float* p, float v) {
    return __hip_atomic_fetch_add(p, v, __ATOMIC_RELAXED, __HIP_MEMORY_SCOPE_AGENT);
}

// ---------------------------------------------------------------- degree ----
__global__ void k_deg_init(float* __restrict__ deg, int n) {
    int i = blockIdx.x * blockDim.x + threadIdx.x;
    if (i < n) deg[i] = 1.0f;  // self-loop weight
}

__global__ void k_deg_edges(const int* __restrict__ dst, const float* __restrict__ w,
                            float* __restrict__ deg, int E) {
    int e = blockIdx.x * blockDim.x + threadIdx.x;
    if (e < E) atomicAddF(&deg[dst[e]], w[e]);
}

// dis = rsqrt(deg); agg1 = dis^2 * x   (self-loop term of layer-1 scalar agg)
__global__ void k_node_prep(const float* __restrict__ deg, const float* __restrict__ x,
                            float* __restrict__ dis, float* __restrict__ agg1, int n) {
    int i = blockIdx.x * blockDim.x + threadIdx.x;
    if (i < n) {
        float d = rsqrtf(deg[i]);   // deg >= 1 always
        dis[i]  = d;
        agg1[i] = d * d * x[i];
    }
}

// layer-1 aggregation on SCALARS (feature dim is 1 before the 1->128 projection)
__global__ void k_agg1_edges(const int* __restrict__ src, const int* __restrict__ dst,
                             const float* __restrict__ w, const float* __restrict__ dis,
                             const float* __restrict__ x, float* __restrict__ agg1, int E) {
    int e = blockIdx.x * blockDim.x + threadIdx.x;
    if (e < E) {
        int s = src[e], d = dst[e];
        atomicAddF(&agg1[d], dis[s] * w[e] * dis[d] * x[s]);
    }
}

// h1[n,f] = relu(agg1[n]*W1[f] + b1[f]); agg2 init with self-loop term dis^2*h1
__global__ void k_h1_expand(const float* __restrict__ agg1, const float* __restrict__ dis,
                            const float* __restrict__ W1, const float* __restrict__ b1,
                            float* __restrict__ h1, float* __restrict__ agg2, int n) {
    int node = blockIdx.x;
    int f = threadIdx.x;                 // 128 threads
    if (node >= n) return;
    float v = fmaxf(agg1[node] * W1[f] + b1[f], 0.0f);
    float d = dis[node];
    int idx = node * 128 + f;
    h1[idx]   = v;
    agg2[idx] = d * d * v;
}

// layer-2 aggregation at width 128 (before the 128->256 GEMM; A(h W) = (A h) W)
// one wave32 per edge, float4 per lane
__global__ void k_agg2_edges(const int* __restrict__ src, const int* __restrict__ dst,
                             const float* __restrict__ w, const float* __restrict__ dis,
                             const float* __restrict__ h1, float* __restrict__ agg2, int E) {
    int e = blockIdx.x * (blockDim.x >> 5) + (threadIdx.x >> 5);
    if (e >= E) return;
    int lane = threadIdx.x & 31;
    int s = src[e], d = dst[e];
    float norm = dis[s] * w[e] * dis[d];
    const float4* hv = (const float4*)(h1 + (size_t)s * 128);
    float4 v = hv[lane];
    float* out = agg2 + (size_t)d * 128 + lane * 4;
    atomicAddF(out + 0, norm * v.x);
    atomicAddF(out + 1, norm * v.y);
    atomicAddF(out + 2, norm * v.z);
    atomicAddF(out + 3, norm * v.w);
}

// ------------------------------------------------------------ WMMA GEMM ----
// C[M x NCOL] = relu(A[M x K] @ B[K x NCOL] + bias), f32 via V_WMMA_F32_16X16X4_F32.
// Block: one 16-row M slab, NCOL/16 waves, each wave one 16x16 output tile.
template <int K, int NCOL>
__global__ __launch_bounds__(32 * (NCOL / 16))
void k_gemm_bias_relu(const float* __restrict__ A, const float* __restrict__ B,
                      const float* __restrict__ bias, float* __restrict__ C, int M) {
    constexpr int NW = NCOL / 16;
    constexpr int LDA = K + 4;                 // stride % 64 == 4 -> conflict-free
    __shared__ float As[16][LDA];

    const int m0  = blockIdx.x * 16;
    const int tid = threadIdx.x;

    // cooperative A-slab load (16 x K)
    for (int i = tid; i < 16 * K; i += 32 * NW) {
        int r = i / K, c = i % K;
        int row = m0 + r;
        As[r][c] = (row < M) ? A[row * K + c] : 0.0f;
    }
    __syncthreads();

    const int wave = tid >> 5;
    const int lane = tid & 31;
    const int r    = lane & 15;   // row (A) / col (B,C) within tile
    const int half = lane >> 4;   // 0: lanes 0-15, 1: lanes 16-31
    const int n0   = wave * 16;

    v8f acc = {};
    for (int k0 = 0; k0 < K; k0 += 4) {
        // A 16x4: vgpr v holds K = 2*half + v  (ISA 7.12.2, 32-bit A layout)
        v2f a = *(const v2f*)&As[r][k0 + 2 * half];
        // B 4x16: vgpr v holds row K = 2*half + v, N striped over lanes
        const float* Bp = B + (k0 + 2 * half) * NCOL + n0 + r;
        v2f b;
        b.x = Bp[0];
        b.y = Bp[NCOL];
        acc = __builtin_amdgcn_wmma_f32_16x16x4_f32(
            /*neg_a=*/false, a, /*neg_b=*/false, b,
            /*c_mod=*/(short)0, acc, /*reuse_a=*/false, /*reuse_b=*/false);
    }

    const float bv = bias[n0 + r];
#pragma unroll
    for (int v = 0; v < 8; ++v) {
        int row = m0 + v + 8 * half;   // C: vgpr v -> M = v + 8*half
        if (row < M) {
            float val = fmaxf(acc[v] + bv, 0.0f);
            C[row * NCOL + n0 + r] = val;
        }
    }
}

// out[n] = dot(h3[n,:128], Wl2) + bl2 ; one wave per node
__global__ void k_final_dot(const float* __restrict__ h3, const float* __restrict__ Wl2,
                            const float* __restrict__ bl2, float* __restrict__ out, int n) {
    int node = blockIdx.x * (blockDim.x >> 5) + (threadIdx.x >> 5);
    if (node >= n) return;
    int lane = threadIdx.x & 31;
    const float4* hv = (const float4*)(h3 + (size_t)node * 128);
    const float4* wv = (const float4*)Wl2;
    float4 a = hv[lane];
    float4 b = wv[lane];
    float s = a.x * b.x + a.y * b.y + a.z * b.z + a.w * b.w;
#pragma unroll
    for (int off = 16; off > 0; off >>= 1) s += __shfl_down(s, off, 32);
    if (lane == 0) out[node] = s + bl2[0];
}

// ----------------------------------------------------------------- launch ---
extern "C" void kernel_launch(void* const* d_in, const int* in_sizes, int n_in,
                              void* d_out, int out_size, void* d_ws, size_t ws_size,
                              hipStream_t stream) {
    const float* x   = (const float*)d_in[0];
    const int*   ei  = (const int*)d_in[1];
    const float* ew  = (const float*)d_in[2];
    const float* W1  = (const float*)d_in[3];
    const float* b1  = (const float*)d_in[4];
    const float* W2  = (const float*)d_in[5];
    const float* b2  = (const float*)d_in[6];
    const float* Wl1 = (const float*)d_in[7];
    const float* bl1 = (const float*)d_in[8];
    const float* Wl2 = (const float*)d_in[9];
    const float* bl2 = (const float*)d_in[10];

    const int N = in_sizes[0];          // 100000
    const int E = in_sizes[2];          // 1600000
    const int* src = ei;
    const int* dst = ei + E;

    // workspace carve-out (512B aligned slices)
    size_t off = 0;
    auto carve = [&](size_t bytes) -> float* {
        float* p = (float*)((char*)d_ws + off);
        off += (bytes + 511) & ~(size_t)511;
        return p;
    };
    float* dis  = carve((size_t)N * 4);
    float* agg1 = carve((size_t)N * 4);
    float* h1   = carve((size_t)N * 128 * 4);   // reused as h3 later
    float* agg2 = carve((size_t)N * 128 * 4);
    float* h2   = carve((size_t)N * 256 * 4);
    float* h3   = h1;                            // h1 dead after k_agg2_edges
    float* deg  = dis;                           // in-place rsqrt

    float* out = (float*)d_out;

    // 1) degrees with self loops
    k_deg_init<<<(N + 255) / 256, 256, 0, stream>>>(deg, N);
    k_deg_edges<<<(E + 255) / 256, 256, 0, stream>>>(dst, ew, deg, E);
    // 2) dis + layer-1 scalar agg (self-loop term)
    k_node_prep<<<(N + 255) / 256, 256, 0, stream>>>(deg, x, dis, agg1, N);
    // 3) layer-1 scalar edge aggregation
    k_agg1_edges<<<(E + 255) / 256, 256, 0, stream>>>(src, dst, ew, dis, x, agg1, E);
    // 4) h1 = relu(agg1*W1 + b1), agg2 = dis^2 * h1
    k_h1_expand<<<N, 128, 0, stream>>>(agg1, dis, W1, b1, h1, agg2, N);
    // 5) layer-2 width-128 edge aggregation (1 wave / edge)
    k_agg2_edges<<<(E + 7) / 8, 256, 0, stream>>>(src, dst, ew, dis, h1, agg2, E);
    // 6) h2 = relu(agg2 @ W2 + b2)     [100000 x 128] @ [128 x 256]
    k_gemm_bias_relu<128, 256><<<(N + 15) / 16, 512, 0, stream>>>(agg2, W2, b2, h2, N);
    // 7) h3 = relu(h2 @ Wl1 + bl1)     [100000 x 256] @ [256 x 128]
    k_gemm_bias_relu<256, 128><<<(N + 15) / 16, 256, 0, stream>>>(h2, Wl1, bl1, h3, N);
    // 8) out = h3 @ Wl2 + bl2          per-node wave dot product
    k_final_dot<<<(N + 7) / 8, 256, 0, stream>>>(h3, Wl2, bl2, out, N);
}